// HodgkinHuxley_69114613727791
// MI455X (gfx1250) — compile-verified
//
#include <hip/hip_runtime.h>
#include <math.h>

// ---------------------------------------------------------------------------
// Hodgkin-Huxley: 10 Euler steps, elementwise over B*C = 4.19M neurons.
// Transcendental-bound => minimize v_exp/v_rcp count, use v_tanh_f32 sigmoid.
// Sigmoid is forced onto v_tanh_f32 (builtin if declared, inline asm else) so
// the 1-trans-per-sigmoid path is guaranteed rather than hoped for.
// ---------------------------------------------------------------------------

static __device__ __forceinline__ float fexp(float x) {
#if __has_builtin(__builtin_amdgcn_exp2f)
    return __builtin_amdgcn_exp2f(x * 1.44269504088896340736f);  // v_exp_f32
#else
    return __expf(x);
#endif
}

static __device__ __forceinline__ float frcp(float x) {
#if __has_builtin(__builtin_amdgcn_rcpf)
    return __builtin_amdgcn_rcpf(x);                             // v_rcp_f32
#else
    return 1.0f / x;
#endif
}

static __device__ __forceinline__ float ftanh(float x) {
#if __has_builtin(__builtin_amdgcn_tanhf)
    return __builtin_amdgcn_tanhf(x);                            // v_tanh_f32
#else
    float r;
    asm("v_tanh_f32 %0, %1" : "=v"(r) : "v"(x));                 // v_tanh_f32
    return r;
#endif
}

static __device__ __forceinline__ float fsig(float x) {
    // sigmoid(x) = 0.5 + 0.5*tanh(x/2): one trans op
    return fmaf(0.5f, ftanh(0.5f * x), 0.5f);
}

#define HH_H      0.002f     // DT / N_STEPS
#define HH_GEPS   1e-5f
#define HH_EM2    0.13533528323661270f  // exp(-2): e55 = e35 * exp(-2)

__global__ __launch_bounds__(256) void hh_euler_kernel(
    const float* __restrict__ state,   // [B, 4C]
    const float* __restrict__ iext,    // [C]
    float* __restrict__ out,           // [B, 4C]
    int nquads,                        // B * C / 4
    int qprShift)                      // log2(C/4); C/4 is a power of two
{
    const int stride = (int)(gridDim.x * blockDim.x);
    const int qpr   = 1 << qprShift;       // quads per component
    const int qmask = qpr - 1;
    const int rowF  = qpr << 4;            // floats per row (4C)
    const int compF = qpr << 2;            // floats per component (C)

    for (int t = blockIdx.x * blockDim.x + threadIdx.x; t < nquads; t += stride) {
        // Prefetch next grid-stride iteration's cache lines (global_prefetch_b8).
        int tn = t + stride;
        if (tn < nquads) {
            const float* pn = state + ((size_t)(tn >> qprShift) * rowF)
                                    + ((size_t)(tn & qmask) << 2);
            __builtin_prefetch(pn,             0, 1);
            __builtin_prefetch(pn + compF,     0, 1);
            __builtin_prefetch(pn + 2 * compF, 0, 1);
            __builtin_prefetch(pn + 3 * compF, 0, 1);
        }

        const int b = t >> qprShift;
        const int q = t & qmask;
        const float4* row = reinterpret_cast<const float4*>(state + (size_t)b * rowF);

        float4 V  = row[q];                 // global_load_b128, coalesced
        float4 LN = row[qpr + q];
        float4 LM = row[2 * qpr + q];
        float4 LH = row[3 * qpr + q];
        float4 IE = reinterpret_cast<const float4*>(iext)[q];

        float vs[4] = {V.x, V.y, V.z, V.w};
        float ln[4] = {LN.x, LN.y, LN.z, LN.w};
        float lm[4] = {LM.x, LM.y, LM.z, LM.w};
        float lh[4] = {LH.x, LH.y, LH.z, LH.w};
        float ie[4] = {IE.x, IE.y, IE.z, IE.w};

        for (int s = 0; s < 10; ++s) {
#pragma unroll
            for (int e = 0; e < 4; ++e) {
                float v = 10.0f * vs[e];

                float n  = fminf(fmaxf(fsig(ln[e]), HH_GEPS), 1.0f - HH_GEPS);
                float m  = fminf(fmaxf(fsig(lm[e]), HH_GEPS), 1.0f - HH_GEPS);
                float hg = fminf(fmaxf(fsig(lh[e]), HH_GEPS), 1.0f - HH_GEPS);

                // Ionic currents
                float n2  = n * n;
                float ina = 120.0f * (m * m * m) * hg * (v - 50.0f);
                float ik  = 36.0f * (n2 * n2) * (v + 77.0f);
                float il  = 0.3f * (v + 54.387f);
                float dvs = (ie[e] - ina - ik - il) * 0.1f;

                // Shared exponentials: e55 = e35 * exp(-2)
                float e35 = fexp(-0.1f * (v + 35.0f));
                float e55 = e35 * HH_EM2;

                float xm = v + 35.0f;
                float xn = v + 55.0f;
                float am = (fabsf(xm) < 0.1f) ? fmaf(0.05f, xm, 1.0f)
                                              : 0.1f  * xm * frcp(1.0f - e35);
                float an = (fabsf(xn) < 0.1f) ? fmaf(0.005f, xn, 0.1f)
                                              : 0.01f * xn * frcp(1.0f - e55);
                float bm = 4.0f   * fexp(-(v + 65.0f) * (1.0f / 18.0f));
                float bn = 0.125f * fexp(-(v + 65.0f) * 0.0125f);
                float ah = 0.07f  * fexp(-(v + 50.0f) * 0.05f);
                float bh = frcp(e35 + 1.0f);   // reuses e35 (no extra exp)

                float dn = an - (an + bn) * n;
                float dm = am - (am + bm) * m;
                float dh = ah - (ah + bh) * hg;

                // Chain rule into logit space: d logit(x) = dx / (x (1-x))
                ln[e] = fmaf(HH_H, dn * frcp(n  * (1.0f - n )), ln[e]);
                lm[e] = fmaf(HH_H, dm * frcp(m  * (1.0f - m )), lm[e]);
                lh[e] = fmaf(HH_H, dh * frcp(hg * (1.0f - hg)), lh[e]);
                vs[e] = fmaf(HH_H, dvs, vs[e]);
            }
        }

        float4* orow = reinterpret_cast<float4*>(out + (size_t)b * rowF);
        orow[q]           = make_float4(vs[0], vs[1], vs[2], vs[3]);
        orow[qpr + q]     = make_float4(ln[0], ln[1], ln[2], ln[3]);
        orow[2 * qpr + q] = make_float4(lm[0], lm[1], lm[2], lm[3]);
        orow[3 * qpr + q] = make_float4(lh[0], lh[1], lh[2], lh[3]);
    }
}

extern "C" void kernel_launch(void* const* d_in, const int* in_sizes, int n_in,
                              void* d_out, int out_size, void* d_ws, size_t ws_size,
                              hipStream_t stream) {
    const float* state = (const float*)d_in[0];   // [B, 4C] fp32
    const float* iext  = (const float*)d_in[1];   // [C] fp32
    float* out = (float*)d_out;

    const int total  = in_sizes[0];   // B * 4C
    const int C      = in_sizes[1];   // 128
    const int nquads = total / 16;    // B * C / 4
    const int qpr    = C / 4;         // 32: power of two

    // log2(qpr) without assuming intrinsics on host
    int qprShift = 0;
    while ((1 << qprShift) < qpr) ++qprShift;

    const int threads = 256;
    int blocks = 1024;                // grid-stride: ~4 quads/thread => prefetch wins
    long long needed = ((long long)nquads + threads - 1) / threads;
    if (needed < blocks) blocks = (int)(needed > 0 ? needed : 1);

    hh_euler_kernel<<<blocks, threads, 0, stream>>>(state, iext, out, nquads, qprShift);
}